// MaxMarginContrastiveLoss_33449205301316
// MI455X (gfx1250) — compile-verified
//
#include <hip/hip_runtime.h>
#include <hip/hip_bf16.h>
#include <math.h>

typedef float v2f __attribute__((ext_vector_type(2)));
typedef float v8f __attribute__((ext_vector_type(8)));

#define DIM 128
#define BR 64              // rows per workgroup (4 waves x 16)
#define BC 128             // columns per LDS tile
#define COLCHUNK 1024      // columns per workgroup (8 tiles)
#define LDS_STRIDE (DIM + 4)
#define MARGIN_F 1.0f

// ---------------- Kernel 1: prep (sq norms, init per-row min/max, zero out) --
__global__ void prep_kernel(const float* __restrict__ X,
                            float* __restrict__ sq,
                            unsigned* __restrict__ minU,
                            unsigned* __restrict__ maxU,
                            float* __restrict__ out, int n) {
    int i = blockIdx.x * blockDim.x + threadIdx.x;
    if (i == 0) out[0] = 0.0f;
    if (i < n) {
        const float* row = X + (size_t)i * DIM;
        float s = 0.0f;
        #pragma unroll 4
        for (int k = 0; k < DIM; ++k) { float v = row[k]; s += v * v; }
        sq[i]   = s;
        minU[i] = 0x7F800000u;   // +inf
        maxU[i] = 0u;            // +0.0f (all dists >= 0)
    }
}

// ---------------- Kernel 2: fused Gram (fp32 WMMA) + masked row min/max ------
__launch_bounds__(128)
__global__ void gram_kernel(const float* __restrict__ X,
                            const int* __restrict__ labels,
                            const float* __restrict__ sq,
                            unsigned* __restrict__ minU,
                            unsigned* __restrict__ maxU, int n) {
    __shared__ float As[BR][LDS_STRIDE];
    __shared__ float Bs[2][BC][LDS_STRIDE];
    __shared__ float sqA[BR];
    __shared__ float sqB[2][BC];
    __shared__ int   labA[BR];
    __shared__ int   labB[2][BC];

    const int tid  = threadIdx.x;      // 0..127
    const int wave = tid >> 5;
    const int lane = tid & 31;
    const int half = lane >> 4;
    const int l15  = lane & 15;
    const int rowBase = blockIdx.x * BR;
    const int cb0 = blockIdx.y * COLCHUNK;
    const int cb1 = cb0 + COLCHUNK;
    const int r16 = wave * 16;

    // Row strip of X into LDS (once).
    for (int i = tid; i < BR * DIM; i += 128) {
        int r = i >> 7, c = i & (DIM - 1);
        As[r][c] = X[(size_t)(rowBase + r) * DIM + c];
    }
    for (int i = tid; i < BR; i += 128) {
        sqA[i]  = sq[rowBase + i];
        labA[i] = labels[rowBase + i];
    }

    // Async-copy one 128x128 B tile (64KB) into LDS buffer `buf`,
    // preserving the padded layout (row stride 528B, 16B-aligned chunks).
    auto issue_tile = [&](int cb, int buf) {
        unsigned ldsBase = (unsigned)(uintptr_t)(&Bs[buf][0][0]);
        const float* src = X + (size_t)cb * DIM;
        #pragma unroll
        for (int it = 0; it < (BC * DIM) / (4 * 128); ++it) {   // 32 chunks/thread
            int idx = it * 128 + tid;          // 16B-chunk id within tile
            int r   = idx >> 5;                // 0..127 (32 chunks per row)
            int cg  = idx & 31;
            unsigned laddr = ldsBase + (unsigned)(r * (LDS_STRIDE * 4) + cg * 16);
            unsigned long long gaddr =
                (unsigned long long)(src + (size_t)r * DIM + cg * 4);
            asm volatile("global_load_async_to_lds_b128 %0, %1, off"
                         :: "v"(laddr), "v"(gaddr) : "memory");
        }
    };
    auto load_meta = [&](int cb, int buf) {
        for (int i = tid; i < BC; i += 128) {
            sqB[buf][i]  = sq[cb + i];
            labB[buf][i] = labels[cb + i];
        }
    };

    int p = 0;
    issue_tile(cb0, 0);
    load_meta(cb0, 0);
    asm volatile("s_wait_asynccnt 0" ::: "memory");
    __syncthreads();   // first tile (and As) visible to all waves

    float minPos[8], maxNeg[8];
    #pragma unroll
    for (int v = 0; v < 8; ++v) { minPos[v] = INFINITY; maxNeg[v] = -INFINITY; }

    for (int cb = cb0; cb < cb1; cb += BC) {
        const int nxt = cb + BC;
        if (nxt < cb1) {                 // overlap next tile copy with compute
            issue_tile(nxt, 1 - p);
            load_meta(nxt, 1 - p);
        }

        v8f acc[8];
        const v8f vzero = {0.f, 0.f, 0.f, 0.f, 0.f, 0.f, 0.f, 0.f};
        #pragma unroll
        for (int j = 0; j < 8; ++j) acc[j] = vzero;

        // 32 x v_wmma_f32_16x16x4_f32 per subtile column group.
        // Frag layout (A==B gather, both come from X):
        //   lanes 0-15  -> row l15, elements {k0, k0+1}
        //   lanes 16-31 -> row l15, elements {k0+2, k0+3}
        #pragma unroll 4
        for (int k0 = 0; k0 < DIM; k0 += 4) {
            const int kk = k0 + 2 * half;
            v2f a;
            a.x = As[r16 + l15][kk];
            a.y = As[r16 + l15][kk + 1];
            #pragma unroll
            for (int j = 0; j < 8; ++j) {
                const int c = j * 16 + l15;
                v2f b;
                b.x = Bs[p][c][kk];
                b.y = Bs[p][c][kk + 1];
                acc[j] = __builtin_amdgcn_wmma_f32_16x16x4_f32(
                    false, a, false, b, (short)0, acc[j], false, false);
            }
        }

        // Epilogue: dot -> distance -> masked min/max.
        // C layout: VGPR v, lanes 0-15 -> M=v, N=lane; lanes 16-31 -> M=v+8.
        #pragma unroll
        for (int j = 0; j < 8; ++j) {
            const int c   = j * 16 + l15;
            const int gc  = cb + c;
            const float sqc = sqB[p][c];
            const int   lc  = labB[p][c];
            #pragma unroll
            for (int v = 0; v < 8; ++v) {
                const int m  = r16 + v + 8 * half;
                const int gr = rowBase + m;
                float d2 = sqA[m] + sqc - 2.0f * acc[j][v];
                d2 = fmaxf(d2, 0.0f);
                const float dist = (d2 > 0.0f) ? sqrtf(d2) : 0.0f;
                const bool same = (labA[m] == lc);
                if (same && (gr != gc)) minPos[v] = fminf(minPos[v], dist);
                if (!same)              maxNeg[v] = fmaxf(maxNeg[v], dist);
            }
        }

        asm volatile("s_wait_asynccnt 0" ::: "memory");  // own async portion done
        __syncthreads();                                  // everyone's portion done
        p ^= 1;
    }

    // Reduce across the 16 lanes of each half-wave (xor masks stay in-half).
    #pragma unroll
    for (int v = 0; v < 8; ++v) {
        #pragma unroll
        for (int off = 1; off < 16; off <<= 1) {
            minPos[v] = fminf(minPos[v], __shfl_xor(minPos[v], off, 32));
            maxNeg[v] = fmaxf(maxNeg[v], __shfl_xor(maxNeg[v], off, 32));
        }
    }

    // Lanes 0 / 16 own rows r16+v / r16+v+8. Commit partials with
    // order-preserving uint atomics (valid since all candidates >= 0).
    if (l15 == 0) {
        #pragma unroll
        for (int v = 0; v < 8; ++v) {
            const int gr = rowBase + r16 + v + 8 * half;
            atomicMin(&minU[gr], __float_as_uint(minPos[v]));
            if (maxNeg[v] >= 0.0f)
                atomicMax(&maxU[gr], __float_as_uint(maxNeg[v]));
        }
    }
}

// ---------------- Kernel 3: fold per-row results into the scalar loss --------
__global__ void finish_kernel(const unsigned* __restrict__ minU,
                              const unsigned* __restrict__ maxU,
                              float* __restrict__ out, int n) {
    int i = blockIdx.x * blockDim.x + threadIdx.x;
    if (i < n) {
        const float pos = __uint_as_float(minU[i]);
        const float neg = __uint_as_float(maxU[i]);
        const float l = fmaxf(pos - neg + MARGIN_F, 0.0f);
        atomicAdd(out, l / (float)n);
    }
}

extern "C" void kernel_launch(void* const* d_in, const int* in_sizes, int n_in,
                              void* d_out, int out_size, void* d_ws, size_t ws_size,
                              hipStream_t stream) {
    const float* X      = (const float*)d_in[0];
    const int*   labels = (const int*)d_in[1];
    const int n = in_sizes[0] / DIM;           // 8192

    float*    sqnorm = (float*)d_ws;           // n floats
    unsigned* minU   = (unsigned*)d_ws + n;    // n uints
    unsigned* maxU   = (unsigned*)d_ws + 2 * n;
    float*    out    = (float*)d_out;

    prep_kernel<<<(n + 255) / 256, 256, 0, stream>>>(X, sqnorm, minU, maxU, out, n);

    dim3 grid(n / BR, n / COLCHUNK);           // 128 x 8 = 1024 blocks
    gram_kernel<<<grid, 128, 0, stream>>>(X, labels, sqnorm, minU, maxU, n);

    finish_kernel<<<(n + 255) / 256, 256, 0, stream>>>(minU, maxU, out, n);
}